// InterpretableMultiHeadAttention_69587060129833
// MI455X (gfx1250) — compile-verified
//
#include <hip/hip_runtime.h>
#include <hip/hip_bf16.h>
#include <cstddef>
#include <cstdint>

// Problem constants (from reference)
#define NH 8
#define BB 4
#define SS 2048
#define DM 512
#define DKK 64

typedef __attribute__((ext_vector_type(2))) float v2f;
typedef __attribute__((ext_vector_type(8))) float v8f;
typedef __attribute__((ext_vector_type(4))) int v4i;

// D = A(16x4 f32) * B(4x16 f32) + C(16x16 f32), wave32 (codegen-confirmed round 1).
__device__ __forceinline__ v8f wmma_f32_4(v2f a, v2f b, v8f c) {
  return __builtin_amdgcn_wmma_f32_16x16x4_f32(false, a, false, b, (short)0, c,
                                               false, false);
}

// CDNA5 async global->LDS copy path (ASYNCcnt) with guarded fallback.
// Builtin signature (from round-2 diagnostic): (v4i addrspace(1)*, v4i addrspace(3)*, imm, imm)
#if __has_builtin(__builtin_amdgcn_global_load_async_to_lds_b128) && \
    __has_builtin(__builtin_amdgcn_s_wait_asynccnt)
#define USE_ASYNC_LDS 1
typedef __attribute__((address_space(1))) v4i* g4_ptr;
typedef __attribute__((address_space(3))) v4i* l4_ptr;
#endif

// ---------------------------------------------------------------------------
// Kernel 1: projections.  qs[h,b,s,64] = q[b,s,:] @ Wq[h]; ks likewise; vs = v @ Wv.
// One block = one (matrix, h, b, 16-row s-tile); 128 threads = 4 waves,
// each wave computes one 16-wide N-tile with 128 K-chunks of f32 WMMA.
// ---------------------------------------------------------------------------
__global__ __launch_bounds__(128) void mha_proj_kernel(
    const float* __restrict__ q, const float* __restrict__ k,
    const float* __restrict__ v, const float* __restrict__ Wq,
    const float* __restrict__ Wk, const float* __restrict__ Wv,
    float* __restrict__ qs, float* __restrict__ ks, float* __restrict__ vs) {
  __shared__ float tile[16 * 516];  // padded pitch: bank = 4*M + k (conflict-free)

  int id = blockIdx.x;
  const float* src;
  const float* W;
  float* dst;
  int b, st;
  if (id < NH * BB * (SS / 16)) {                       // qs blocks
    int h = id >> 9; int r = id & 511; b = r >> 7; st = r & 127;
    src = q; W = Wq + (size_t)h * DM * DKK;
    dst = qs + ((size_t)(h * BB + b) * SS + st * 16) * DKK;
  } else if (id < 2 * NH * BB * (SS / 16)) {            // ks blocks
    id -= NH * BB * (SS / 16);
    int h = id >> 9; int r = id & 511; b = r >> 7; st = r & 127;
    src = k; W = Wk + (size_t)h * DM * DKK;
    dst = ks + ((size_t)(h * BB + b) * SS + st * 16) * DKK;
  } else {                                              // vs blocks
    id -= 2 * NH * BB * (SS / 16);
    b = id >> 7; st = id & 127;
    src = v; W = Wv;
    dst = vs + ((size_t)b * SS + st * 16) * DKK;
  }

  // Stage 16 rows x 512 cols into LDS.
  const float* rowbase = src + ((size_t)b * SS + st * 16) * DM;
#ifdef USE_ASYNC_LDS
  for (int i = threadIdx.x; i < 16 * 128; i += 128) {
    int r = i >> 7, c4 = i & 127;
    g4_ptr g = (g4_ptr)(uintptr_t)(rowbase + (size_t)r * DM + c4 * 4);
    // LDS byte offset = low 32 bits of generic shared address.
    l4_ptr l = (l4_ptr)(unsigned)(uintptr_t)(tile + r * 516 + c4 * 4);
    __builtin_amdgcn_global_load_async_to_lds_b128(g, l, 0, 0);
  }
  __builtin_amdgcn_s_wait_asynccnt(0);
#else
  for (int i = threadIdx.x; i < 16 * 128; i += 128) {
    int r = i >> 7, c4 = i & 127;
    float4 val = ((const float4*)(rowbase + (size_t)r * DM))[c4];
    *((float4*)(tile + r * 516 + c4 * 4)) = val;
  }
#endif
  __syncthreads();

  int wv = threadIdx.x >> 5, lane = threadIdx.x & 31;
  int n0 = wv * 16;
  int MN = lane & 15;                  // M for A / N for B,C
  int koff = (lane < 16) ? 0 : 2;
  int khalf = koff >> 1;

  const v2f* arow = (const v2f*)(tile + MN * 516);  // 8B-aligned (516 even)
  v8f acc = {};
  for (int c = 0; c < 128; ++c) {
    int k0 = 4 * c + koff;
    v2f a = arow[2 * c + khalf];
    v2f bv; bv.x = W[(size_t)k0 * DKK + n0 + MN];
            bv.y = W[(size_t)(k0 + 1) * DKK + n0 + MN];
    acc = wmma_f32_4(a, bv, acc);
  }
  int mb = (lane >> 4) * 8;
  for (int r = 0; r < 8; ++r) dst[(size_t)(mb + r) * DKK + n0 + MN] = acc[r];
}

// ---------------------------------------------------------------------------
// Kernel 2: fused attention per (h*B+b, 16-row q-tile).
// Score strip 16x2048 kept in LDS (pitch 2052 -> conflict-free WMMA A reads).
// 8 waves: score tiles strided over s; softmax per row; P@V with 2-way K split.
// ---------------------------------------------------------------------------
__global__ __launch_bounds__(256) void mha_attn_kernel(
    const float* __restrict__ qs, const float* __restrict__ ks,
    const float* __restrict__ vs, float* __restrict__ attn_out,
    float* __restrict__ heads_ws) {
  extern __shared__ float lds[];
  float* scores = lds;                 // 16 x 2052 (row pitch 8208B, 16B aligned)
  float* hacc = lds + 16 * 2052;       // 16 x 64 f32 head accumulator

  int id = blockIdx.x;
  int st = id & 127;
  int hb = id >> 7;                    // h*B + b
  int b = hb & (BB - 1);
  int s0 = st * 16;
  int tid = threadIdx.x, wv = tid >> 5, lane = tid & 31;
  int MN = lane & 15;
  int koff = (lane < 16) ? 0 : 2;
  int khalf = koff >> 1;
  int mb = (lane >> 4) * 8;

  for (int i = tid; i < 16 * 16; i += 256)
    ((float4*)hacc)[i] = make_float4(0.f, 0.f, 0.f, 0.f);

  // Preload Q A-operands (16 K-chunks covering d_k = 64) as b64 loads.
  const v2f* qrow =
      (const v2f*)(qs + ((size_t)hb * SS + s0) * DKK + (size_t)MN * DKK);
  v2f qa[16];
  for (int c = 0; c < 16; ++c) qa[c] = qrow[2 * c + khalf];

  // ---- scores = (Q K^T) * scale, written into LDS strip ----
  const float scale = 0.125f;  // 1/sqrt(64)
  const float* kbase = ks + (size_t)hb * SS * DKK;
  for (int t = wv; t < SS / 16; t += 8) {
    v8f acc = {};
    const v2f* krow = (const v2f*)(kbase + (size_t)(t * 16 + MN) * DKK);
    for (int c = 0; c < 16; ++c)
      acc = wmma_f32_4(qa[c], krow[2 * c + khalf], acc);  // b64 B-operand, L2-hot
    for (int r = 0; r < 8; ++r)
      scores[(mb + r) * 2052 + t * 16 + MN] = acc[r] * scale;
  }
  __syncthreads();

  // ---- softmax over each of the 16 rows (2 rows per wave) + write attn ----
  for (int rr = 0; rr < 2; ++rr) {
    int row = wv * 2 + rr;
    float* sr = scores + row * 2052;
    float4* sr4 = (float4*)sr;
    float mx = -3.402823466e38f;
    for (int j = lane; j < SS / 4; j += 32) {
      float4 x = sr4[j];
      mx = fmaxf(mx, fmaxf(fmaxf(x.x, x.y), fmaxf(x.z, x.w)));
    }
    for (int off = 16; off > 0; off >>= 1) mx = fmaxf(mx, __shfl_xor(mx, off, 32));
    float sum = 0.f;
    for (int j = lane; j < SS / 4; j += 32) {
      float4 x = sr4[j];
      x.x = __expf(x.x - mx); x.y = __expf(x.y - mx);
      x.z = __expf(x.z - mx); x.w = __expf(x.w - mx);
      sr4[j] = x;
      sum += (x.x + x.y) + (x.z + x.w);
    }
    for (int off = 16; off > 0; off >>= 1) sum += __shfl_xor(sum, off, 32);
    float inv = 1.0f / sum;
    float4* arow4 = (float4*)(attn_out + ((size_t)hb * SS + s0 + row) * SS);
    for (int j = lane; j < SS / 4; j += 32) {
      float4 x = sr4[j];
      x.x *= inv; x.y *= inv; x.z *= inv; x.w *= inv;
      sr4[j] = x;
      arow4[j] = x;                    // coalesced b128 stores: 512 MiB attn output
    }
  }
  __syncthreads();

  // ---- heads(16x64) = P(16x2048) @ V(2048x64); 4 N-tiles x 2 K-halves ----
  int n0 = (wv & 3) * 16;
  int kh = wv >> 2;
  const float* vbase = vs + (size_t)b * SS * DKK;
  const v2f* prow = (const v2f*)(scores + MN * 2052);  // 8B aligned (2052 even)
  v8f acc = {};
  for (int c = kh * 256; c < kh * 256 + 256; ++c) {
    int k0 = 4 * c + koff;
    v2f a = prow[2 * c + khalf];       // ds_load_b64, conflict-free by pitch
    v2f bv; bv.x = vbase[(size_t)k0 * DKK + n0 + MN];
            bv.y = vbase[(size_t)(k0 + 1) * DKK + n0 + MN];
    acc = wmma_f32_4(a, bv, acc);
  }
  // Exactly 2 contributions per cell; float add is commutative -> deterministic.
  for (int r = 0; r < 8; ++r) atomicAdd(&hacc[(mb + r) * 64 + n0 + MN], acc[r]);
  __syncthreads();

  float4* hdst = (float4*)(heads_ws + ((size_t)hb * SS + s0) * DKK);
  for (int i = tid; i < 16 * 16; i += 256) hdst[i] = ((float4*)hacc)[i];
}

// ---------------------------------------------------------------------------
// Kernel 3: out = mean_h(heads) @ Wo.  One block per 16-row (b,s) tile.
// ---------------------------------------------------------------------------
__global__ __launch_bounds__(256) void mha_outproj_kernel(
    const float* __restrict__ heads, const float* __restrict__ Wo,
    float* __restrict__ out) {
  __shared__ float mean[16 * 68];      // pitch 68 -> bank = 4*M + k

  int row0 = blockIdx.x * 16;          // flat (b*S + s) row
  int tid = threadIdx.x;
  for (int i = tid; i < 16 * 64; i += 256) {
    int r = i >> 6, c = i & 63;
    float s = 0.f;
    for (int h = 0; h < NH; ++h)
      s += heads[((size_t)h * BB * SS + row0 + r) * DKK + c];
    mean[r * 68 + c] = s * 0.125f;     // /NH
  }
  __syncthreads();

  int wv = tid >> 5, lane = tid & 31;
  int MN = lane & 15;
  int koff = (lane < 16) ? 0 : 2;
  int khalf = koff >> 1;
  int mb = (lane >> 4) * 8;

  const v2f* arow = (const v2f*)(mean + MN * 68);
  v2f a[16];
  for (int c = 0; c < 16; ++c) a[c] = arow[2 * c + khalf];

  for (int nt = wv; nt < DM / 16; nt += 8) {
    int n0 = nt * 16;
    v8f acc = {};
    for (int c = 0; c < 16; ++c) {
      int k0 = 4 * c + koff;
      v2f bv; bv.x = Wo[(size_t)k0 * DM + n0 + MN];
              bv.y = Wo[(size_t)(k0 + 1) * DM + n0 + MN];
      acc = wmma_f32_4(a[c], bv, acc);
    }
    for (int r = 0; r < 8; ++r)
      out[((size_t)row0 + mb + r) * DM + n0 + MN] = acc[r];
  }
}

// ---------------------------------------------------------------------------
extern "C" void kernel_launch(void* const* d_in, const int* in_sizes, int n_in,
                              void* d_out, int out_size, void* d_ws,
                              size_t ws_size, hipStream_t stream) {
  const float* q  = (const float*)d_in[0];
  const float* k  = (const float*)d_in[1];
  const float* v  = (const float*)d_in[2];
  const float* Wq = (const float*)d_in[3];
  const float* Wk = (const float*)d_in[4];
  const float* Wv = (const float*)d_in[5];
  const float* Wo = (const float*)d_in[6];

  float* out      = (float*)d_out;                        // [B,S,512]
  float* attn_out = (float*)d_out + (size_t)BB * SS * DM; // [H,B,S,S]

  // Workspace layout (floats): qs 16MB, ks 16MB, vs 2MB, heads 16MB = 50MB.
  float* ws    = (float*)d_ws;
  float* qs    = ws;
  float* ksb   = ws + (size_t)NH * BB * SS * DKK;          // +4,194,304
  float* vsb   = ksb + (size_t)NH * BB * SS * DKK;         // +4,194,304
  float* heads = vsb + (size_t)BB * SS * DKK;              // +524,288

  // 1) projections: qs(4096) + ks(4096) + vs(512) blocks
  mha_proj_kernel<<<2 * NH * BB * (SS / 16) + BB * (SS / 16), 128, 0, stream>>>(
      q, k, v, Wq, Wk, Wv, qs, ksb, vsb);

  // 2) fused attention: one block per (h,b,q-tile); 128.25KB score strip + 4KB acc
  size_t shmem = (16 * 2052 + 16 * 64) * sizeof(float);
  mha_attn_kernel<<<NH * BB * (SS / 16), 256, shmem, stream>>>(
      qs, ksb, vsb, attn_out, heads);

  // 3) mean over heads + output projection
  mha_outproj_kernel<<<(BB * SS) / 16, 256, 0, stream>>>(heads, Wo, out);
}